// MSDeformAttnPose_19988777796194
// MI455X (gfx1250) — compile-verified
//
#include <hip/hip_runtime.h>
#include <math.h>

typedef float v2f __attribute__((ext_vector_type(2)));
typedef float v8f __attribute__((ext_vector_type(8)));

#define D_MODEL  256
#define N_HEADS  8
#define N_LEVELS 4
#define N_POINTS 4
#define LEN_Q    5100   // 300 * 17
#define BS       4
#define KDIM     256
#define NPROJ    384    // 256 offset outputs + 128 attn logits

// ---------------------------------------------------------------------------
// Kernel 1: proj[M,384] = query[M,256] @ [W_off; W_attn]^T + [b_off; b_attn]
// One wave (32 threads) computes a 16(M) x 64(N) tile using fp32 WMMA.
// A-frag (16x4 f32): lanes 0-15 -> M=lane, K={k0,k0+1}; lanes 16-31 -> K={k0+2,k0+3}
// B-frag (4x16 f32): lanes 0-15 -> N=lane, K={k0,k0+1}; lanes 16-31 -> K={k0+2,k0+3}
// C/D  (16x16 f32): VGPR r: lanes 0-15 -> M=r, lanes 16-31 -> M=r+8
// The W_off / W_attn split is uniform per block (64 | 256), so the weight
// base pointer is selected with a uniform branch -> global_load (not flat).
// ---------------------------------------------------------------------------
__global__ __launch_bounds__(32)
void proj_gemm(const float* __restrict__ Q,
               const float* __restrict__ Woff,  const float* __restrict__ boff,
               const float* __restrict__ Wattn, const float* __restrict__ battn,
               float* __restrict__ proj)
{
    const int lane  = threadIdx.x & 31;
    const int mBase = blockIdx.x * 16;
    const int nBase = blockIdx.y * 64;

    const int ml   = lane & 15;          // A: M row in tile; B: N col in tile
    const int kOff = (lane >> 4) * 2;    // K sub-offset within a K=4 step

    const float* Arow = Q + (size_t)(mBase + ml) * KDIM + kOff;

    // Uniform selection: each 64-wide N block lies entirely in W_off or W_attn.
    const float* Wbase;
    const float* bbase;
    int nLoc;
    if (nBase < 256) { Wbase = Woff;  bbase = boff;  nLoc = nBase; }
    else             { Wbase = Wattn; bbase = battn; nLoc = nBase - 256; }

    const float* Brow[4];
    float bias[4];
#pragma unroll
    for (int nt = 0; nt < 4; ++nt) {
        const int n = nLoc + nt * 16 + ml;
        Brow[nt] = Wbase + (size_t)n * KDIM + kOff;
        bias[nt] = bbase[nLoc + nt * 16 + ml];
    }

    v8f acc[4] = {};

#pragma unroll 4
    for (int k0 = 0; k0 < KDIM; k0 += 4) {
        v2f a = *(const v2f*)(Arow + k0);
#pragma unroll
        for (int nt = 0; nt < 4; ++nt) {
            v2f b = *(const v2f*)(Brow[nt] + k0);
            acc[nt] = __builtin_amdgcn_wmma_f32_16x16x4_f32(
                /*neg_a=*/false, a, /*neg_b=*/false, b,
                /*c_mod=*/(short)0, acc[nt],
                /*reuse_a=*/false, /*reuse_b=*/false);
        }
    }

    const int rowHalf = (lane >> 4) * 8;
#pragma unroll
    for (int nt = 0; nt < 4; ++nt) {
        float* outp = proj + (size_t)(mBase + rowHalf) * NPROJ + nBase + nt * 16 + ml;
#pragma unroll
        for (int r = 0; r < 8; ++r)
            outp[(size_t)r * NPROJ] = acc[nt][r] + bias[nt];
    }
}

// ---------------------------------------------------------------------------
// Kernel 2: softmax + bilinear sampling + weighted sum.
// One wave per (b, h, q); lane = channel (hd == 32 == wave32).
// ---------------------------------------------------------------------------
__global__ __launch_bounds__(128)
void msdeform_sample(const float* __restrict__ proj,
                     const float* __restrict__ refp,
                     const float* __restrict__ v0, const float* __restrict__ v1,
                     const float* __restrict__ v2, const float* __restrict__ v3,
                     float* __restrict__ out)
{
    const int lane = threadIdx.x & 31;
    const int gw   = blockIdx.x * 4 + (threadIdx.x >> 5);
    const int total = BS * N_HEADS * LEN_Q;
    if (gw >= total) return;

    const int n = gw / LEN_Q;            // n = b*H + h  (matches value batch dim)
    const int q = gw - n * LEN_Q;
    const int b = n >> 3;
    const int h = n & 7;

    const float* prow = proj + (size_t)(b * LEN_Q + q) * NPROJ;

    // softmax over the 16 attention logits of this head (lane-redundant scalar math)
    float a[16];
    float amax = -1e30f;
#pragma unroll
    for (int j = 0; j < 16; ++j) {
        a[j] = prow[256 + h * 16 + j];
        amax = fmaxf(amax, a[j]);
    }
    float asum = 0.f;
#pragma unroll
    for (int j = 0; j < 16; ++j) {
        a[j] = __expf(a[j] - amax);
        asum += a[j];
    }
    const float ainv = 1.f / asum;

    // flat q -> (qi, ki): q = qi*17 + ki   (rp transpose(0,1,3,2,4) collapse)
    const int qi = q / 17;
    const int ki = q - qi * 17;

    const float* vals[4] = {v0, v1, v2, v3};
    const int   HWs[4] = {16384, 4096, 1024, 256};
    const int   Hs [4] = {128, 64, 32, 16};
    const int   Ws [4] = {128, 64, 32, 16};

    float acc = 0.f;
#pragma unroll
    for (int l = 0; l < 4; ++l) {
        const int Hh = Hs[l], Ww = Ws[l], HW = HWs[l];
        const float* vbase = vals[l] + (size_t)(n * 32 + lane) * HW;
        const float* rp = refp + (size_t)((((b * 300 + qi) * N_LEVELS) + l) * 17 + ki) * 2;
        const float rpx = rp[0];
        const float rpy = rp[1];
#pragma unroll
        for (int p = 0; p < N_POINTS; ++p) {
            const int oidx = ((h * 4 + l) * 4 + p) * 2;
            const float ox = prow[oidx];
            const float oy = prow[oidx + 1];
            const float wlp = a[l * 4 + p] * ainv;

            // grid math collapses: x = rp_x*W + off_x - 0.5 ; y = rp_y*H + off_y - 0.5
            const float x = rpx * (float)Ww + ox - 0.5f;
            const float y = rpy * (float)Hh + oy - 0.5f;
            const float x0f = floorf(x), y0f = floorf(y);
            const int   x0  = (int)x0f,  y0  = (int)y0f;
            const float fx = x - x0f, fy = y - y0f;

            float s = 0.f;
#pragma unroll
            for (int t = 0; t < 4; ++t) {
                const int xi = x0 + (t & 1);
                const int yi = y0 + (t >> 1);
                const float wt = ((t & 1) ? fx : 1.f - fx) * ((t >> 1) ? fy : 1.f - fy);
                if (xi >= 0 && xi < Ww && yi >= 0 && yi < Hh)
                    s += wt * vbase[yi * Ww + xi];
            }
            acc += wlp * s;
        }
    }

    out[(size_t)(b * LEN_Q + q) * D_MODEL + h * 32 + lane] = acc;
}

// ---------------------------------------------------------------------------
extern "C" void kernel_launch(void* const* d_in, const int* in_sizes, int n_in,
                              void* d_out, int out_size, void* d_ws, size_t ws_size,
                              hipStream_t stream)
{
    const float* query = (const float*)d_in[0];
    const float* refp  = (const float*)d_in[1];
    const float* v0    = (const float*)d_in[2];
    const float* v1    = (const float*)d_in[3];
    const float* v2    = (const float*)d_in[4];
    const float* v3    = (const float*)d_in[5];
    const float* Woff  = (const float*)d_in[6];
    const float* boff  = (const float*)d_in[7];
    const float* Wattn = (const float*)d_in[8];
    const float* battn = (const float*)d_in[9];

    float* out  = (float*)d_out;
    float* proj = (float*)d_ws;   // 20400 * 384 floats = 31.3 MB scratch

    // GEMM: M=20400 (1275 tiles of 16), N=384 (6 tiles of 64), one wave per tile
    dim3 g1(BS * LEN_Q / 16, NPROJ / 64);
    proj_gemm<<<g1, 32, 0, stream>>>(query, Woff, boff, Wattn, battn, proj);

    // Sampling: one wave per (b,h,q); 4 waves per 128-thread block
    const int total  = BS * N_HEADS * LEN_Q;       // 163200 waves
    const int blocks = (total + 3) / 4;
    msdeform_sample<<<blocks, 128, 0, stream>>>(proj, refp, v0, v1, v2, v3, out);
}